// NewAIFV_35639638622594
// MI455X (gfx1250) — compile-verified
//
#include <hip/hip_runtime.h>
#include <hip/hip_bf16.h>
#include <math.h>

#define BB    4
#define CC    256
#define IMGD  160
#define HWD   25600
#define PPD   100
#define NBLKD 256

typedef __attribute__((ext_vector_type(2))) float v2f;
typedef __attribute__((ext_vector_type(8))) float v8f;

// ---- workspace layout (float offsets) ----
#define OFF_SUMTS 0        // [BB][CC][4]  = 4096
#define OFF_SUMTT 4096     // [BB][CC][4]  = 4096
#define OFF_CNT   8192     // [BB][4]      = 16
#define OFF_SCT   8208     // [BB][4][4]   = 64   sumScT[b][j][i]
#define OFF_PS    8272     // [3][BB][4][CC] = 12288
#define OFF_PT    20560    // 12288
#define OFF_PNS   32848    // 48
#define OFF_PNT   32896    // 48
#define OFF_LOSS  32944    // 1
#define STATS_N   32945
#define OFF_W     32960    // [BB*HWD]
#define OFF_SW    135360   // [BB*HWD]
#define OFF_CLS   237760   // [BB*HWD] ints

// gather index: j in [0,25600) -> flat HW position
__device__ __forceinline__ int gidx(int j) {
  int i = j >> 8;       // j / 256
  int o = j & 255;
  return (o >> 4) * IMGD + (o & 15) * 16 + (i >> 4) * (16 * IMGD) + (i & 15);
}

__global__ __launch_bounds__(256) void kzero(float* p, int n) {
  int i = blockIdx.x * 256 + threadIdx.x;
  if (i < n) p[i] = 0.0f;
}

// ---- K1: softmax + gather of predictions, per-pixel class/w/sw + small sums ----
__global__ __launch_bounds__(256) void k1_pixels(const float* __restrict__ PS,
                                                 const float* __restrict__ PT,
                                                 const int*   __restrict__ tgt,
                                                 int* clsbuf, float* wbuf, float* swbuf,
                                                 float* cnt, float* sct) {
  __shared__ float bins[20];   // [0..3]=cnt, [4+j*4+i]=sumScT
  if (threadIdx.x < 20) bins[threadIdx.x] = 0.0f;
  __syncthreads();
  int s = blockIdx.x * 256 + threadIdx.x;     // [0, BB*HWD)
  int b = s / HWD;
  int t = s - b * HWD;
  int g = gidx(t);
  int cls = tgt[b * HWD + g];
  float ps[4], pt[4];
#pragma unroll
  for (int q = 0; q < 4; ++q) {
    ps[q] = PS[(b * 4 + q) * HWD + g];
    pt[q] = PT[(b * 4 + q) * HWD + g];
  }
  float mS = fmaxf(fmaxf(ps[0], ps[1]), fmaxf(ps[2], ps[3]));
  float mT = fmaxf(fmaxf(pt[0], pt[1]), fmaxf(pt[2], pt[3]));
  float sS = 0.0f, sT = 0.0f;
#pragma unroll
  for (int q = 0; q < 4; ++q) {
    ps[q] = __expf(ps[q] - mS); sS += ps[q];
    pt[q] = __expf(pt[q] - mT); sT += pt[q];
  }
  float iS = 1.0f / sS, iT = 1.0f / sT;
#pragma unroll
  for (int q = 0; q < 4; ++q) { ps[q] *= iS; pt[q] *= iT; }
  float sw = pt[cls];
  float w  = (1.0f - ps[cls]) * sw;
  clsbuf[s] = cls; wbuf[s] = w; swbuf[s] = sw;
  atomicAdd(&bins[cls], 1.0f);
#pragma unroll
  for (int q = 0; q < 4; ++q) atomicAdd(&bins[4 + q * 4 + cls], pt[q]);
  __syncthreads();
  if (threadIdx.x < 4)  atomicAdd(&cnt[b * 4 + threadIdx.x], bins[threadIdx.x]);
  if (threadIdx.x >= 4 && threadIdx.x < 20)
    atomicAdd(&sct[b * 16 + (threadIdx.x - 4)], bins[threadIdx.x]);
}

// cooperative stage of both attention matrices, TRANSPOSED: lds[col*100 + k]
__device__ __forceinline__ void stage_atten(float* ldsS, float* ldsT,
                                            const float* __restrict__ ASg,
                                            const float* __restrict__ ATg) {
  for (int idx = threadIdx.x; idx < PPD * PPD; idx += 256) {
    int k   = idx / PPD;
    int col = idx - k * PPD;        // source is [k][col]; dest is [col][k]
    ldsS[col * PPD + k] = ASg[idx]; // coalesced reads, ~coalesced LDS writes
    ldsT[col * PPD + k] = ATg[idx];
  }
  __syncthreads();
}

// ---- K2: WMMA GEMM pass 1 — class-masked sums of TS and sw-weighted TT ----
__global__ __launch_bounds__(256) void k2_stats(const float* __restrict__ fS,
                                                const float* __restrict__ fT,
                                                const float* __restrict__ AS,
                                                const float* __restrict__ AT,
                                                const int*   __restrict__ clsbuf,
                                                const float* __restrict__ swbuf,
                                                float* sumTS, float* sumTTw) {
  __shared__ __align__(16) float ldsS[PPD * PPD];   // 40 KB
  __shared__ __align__(16) float ldsT[PPD * PPD];   // 40 KB
  int bblk = blockIdx.x >> 9;                       // 512 blocks per batch
  stage_atten(ldsS, ldsT, AS + bblk * PPD * PPD, AT + bblk * PPD * PPD);

  int wid  = blockIdx.x * 8 + (threadIdx.x >> 5);   // 16384 waves: b*4096 + c*16 + nt
  int lane = threadIdx.x & 31;
  int b    = wid >> 12;
  int rem  = wid & 4095;
  int c    = rem >> 4;
  int n0   = (rem & 15) << 4;
  int lm   = lane & 15, half = lane >> 4;

  const float* fSb = fS + (size_t)(b * CC + c) * HWD;
  const float* fTb = fT + (size_t)(b * CC + c) * HWD;

  int rowj = (n0 + lm) * PPD;
  v2f aS[25], aT[25];
#pragma unroll
  for (int st = 0; st < 25; ++st) {
    int k  = st * 4 + half * 2;
    int g0 = gidx(rowj + k), g1 = gidx(rowj + k + 1);
    v2f a; a.x = fSb[g0]; a.y = fSb[g1]; aS[st] = a;
    v2f t; t.x = fTb[g0]; t.y = fTb[g1]; aT[st] = t;
  }

  float binS[4] = {0.f, 0.f, 0.f, 0.f};
  float binT[4] = {0.f, 0.f, 0.f, 0.f};

  for (int jt = 0; jt < 7; ++jt) {
    int col   = jt * 16 + lm;
    bool cok  = (col < PPD);
    int  colc = cok ? col : 0;      // clamp: padded columns computed but discarded
    const float* bSrow = &ldsS[colc * PPD];
    const float* bTrow = &ldsT[colc * PPD];
    v8f accS = {}; v8f accT = {};
#pragma unroll
    for (int st = 0; st < 25; ++st) {
      int k = st * 4 + half * 2;
      v2f bS = *(const v2f*)(bSrow + k);   // single ds_load_b64, bank-conflict-free
      v2f bT = *(const v2f*)(bTrow + k);
      accS = __builtin_amdgcn_wmma_f32_16x16x4_f32(false, aS[st], false, bS,
                                                   (short)0, accS, false, false);
      accT = __builtin_amdgcn_wmma_f32_16x16x4_f32(false, aT[st], false, bT,
                                                   (short)0, accT, false, false);
    }
    if (cok) {
#pragma unroll
      for (int v = 0; v < 8; ++v) {
        int n    = n0 + v + half * 8;
        int spix = n * PPD + col;
        int cl   = clsbuf[b * HWD + spix];
        float vs = accS[v];
        float vt = accT[v] * swbuf[b * HWD + spix];
#pragma unroll
        for (int q = 0; q < 4; ++q) {
          binS[q] += (cl == q) ? vs : 0.0f;
          binT[q] += (cl == q) ? vt : 0.0f;
        }
      }
    }
  }
#pragma unroll
  for (int q = 0; q < 4; ++q) {
    float vS = binS[q], vT = binT[q];
    for (int off = 16; off > 0; off >>= 1) {
      vS += __shfl_xor(vS, off, 32);
      vT += __shfl_xor(vT, off, 32);
    }
    if (lane == 0) {
      atomicAdd(&sumTS [(b * CC + c) * 4 + q], vS);
      atomicAdd(&sumTTw[(b * CC + c) * 4 + q], vT);
    }
  }
}

// ---- K3: build 3 prototype variants + their norms ----
__global__ __launch_bounds__(256) void k3_protos(const float* __restrict__ sumTS,
                                                 const float* __restrict__ sumTTw,
                                                 const float* __restrict__ cnt,
                                                 const float* __restrict__ sct,
                                                 float* protoS, float* protoT,
                                                 float* pnS, float* pnT) {
  __shared__ float red[256];
  int blk = blockIdx.x;                   // 3*BB*4 = 48
  int vv  = blk / (BB * 4);
  int rem = blk - vv * (BB * 4);
  int b   = rem >> 2;
  int i   = rem & 3;
  const int CL0[4] = {1, 0, 3, 2}, CL1[4] = {2, 3, 0, 1}, CL2[4] = {3, 2, 1, 0};
  int j = (vv == 0) ? CL0[i] : ((vv == 1) ? CL1[i] : CL2[i]);
  bool prto = (vv != 0);
  float eps = (vv == 2) ? 1.0f : 1e-6f;
  float denS = (prto ? cnt[b * 4 + j] : cnt[b * 4 + i]) + eps;
  float denT = (prto ? sct[b * 16 + j * 4 + j] : sct[b * 16 + j * 4 + i]) + eps;
  int c = threadIdx.x;
  float pSv = sumTS [(b * CC + c) * 4 + j] / denS;
  float pTv = sumTTw[(b * CC + c) * 4 + j] / denT;
  int po = ((vv * BB + b) * 4 + i) * CC + c;
  protoS[po] = pSv; protoT[po] = pTv;
  red[c] = pSv * pSv; __syncthreads();
  for (int st = 128; st > 0; st >>= 1) { if (c < st) red[c] += red[c + st]; __syncthreads(); }
  if (c == 0) pnS[(vv * BB + b) * 4 + i] = sqrtf(red[0]);
  __syncthreads();
  red[c] = pTv * pTv; __syncthreads();
  for (int st = 128; st > 0; st >>= 1) { if (c < st) red[c] += red[c + st]; __syncthreads(); }
  if (c == 0) pnT[(vv * BB + b) * 4 + i] = sqrtf(red[0]);
}

// ---- K4: WMMA GEMM pass 2 — per-pixel cosines vs prototypes, loss ----
__global__ __launch_bounds__(256) void k4_loss(const float* __restrict__ fS,
                                               const float* __restrict__ fT,
                                               const float* __restrict__ AS,
                                               const float* __restrict__ AT,
                                               const int*   __restrict__ clsbuf,
                                               const float* __restrict__ wbuf,
                                               const float* __restrict__ protoS,
                                               const float* __restrict__ protoT,
                                               const float* __restrict__ pnS,
                                               const float* __restrict__ pnT,
                                               float* lossAcc) {
  __shared__ __align__(16) float ldsS[PPD * PPD];   // 40 KB
  __shared__ __align__(16) float ldsT[PPD * PPD];   // 40 KB
  int bblk = blockIdx.x / 224;                      // 224 blocks per batch
  stage_atten(ldsS, ldsT, AS + bblk * PPD * PPD, AT + bblk * PPD * PPD);

  int wid  = blockIdx.x * 8 + (threadIdx.x >> 5);   // 7168 waves: (b*256+n)*7 + jt
  int lane = threadIdx.x & 31;
  int jt   = wid % 7;
  int rem  = wid / 7;
  int n    = rem & 255;
  int b    = rem >> 8;
  int lm   = lane & 15, half = lane >> 4;

  int col   = jt * 16 + lm;
  bool cok  = (col < PPD);
  int  colc = cok ? col : 0;
  int spix  = n * PPD + col;
  int cls   = cok ? clsbuf[b * HWD + spix] : 0;

  // B fragments (fixed per wave) from LDS; gather indices fixed per lane
  v2f bS[25], bT[25];
  int g0a[25], g1a[25];
  int rowj = n * PPD;
#pragma unroll
  for (int st = 0; st < 25; ++st) {
    int k = st * 4 + half * 2;
    bS[st] = *(const v2f*)(&ldsS[colc * PPD] + k);
    bT[st] = *(const v2f*)(&ldsT[colc * PPD] + k);
    g0a[st] = gidx(rowj + k);
    g1a[st] = gidx(rowj + k + 1);
  }

  float dS[3] = {0.f, 0.f, 0.f}, dT[3] = {0.f, 0.f, 0.f};
  float ns2 = 0.0f, nt2 = 0.0f;

  for (int ct = 0; ct < 16; ++ct) {
    int c0 = ct * 16;
    const float* fSb = fS + (size_t)(b * CC + c0 + lm) * HWD;
    const float* fTb = fT + (size_t)(b * CC + c0 + lm) * HWD;
    v8f accS = {}; v8f accT = {};
#pragma unroll
    for (int st = 0; st < 25; ++st) {
      v2f a; a.x = fSb[g0a[st]]; a.y = fSb[g1a[st]];
      v2f t; t.x = fTb[g0a[st]]; t.y = fTb[g1a[st]];
      accS = __builtin_amdgcn_wmma_f32_16x16x4_f32(false, a, false, bS[st],
                                                   (short)0, accS, false, false);
      accT = __builtin_amdgcn_wmma_f32_16x16x4_f32(false, t, false, bT[st],
                                                   (short)0, accT, false, false);
    }
#pragma unroll
    for (int v = 0; v < 8; ++v) {
      int c = c0 + v + half * 8;
      float ts = accS[v], tt = accT[v];
      ns2 += ts * ts; nt2 += tt * tt;
#pragma unroll
      for (int vv = 0; vv < 3; ++vv) {
        size_t off = ((size_t)(vv * BB + b) * 4 + cls) * CC + c;
        dS[vv] += ts * protoS[off];
        dT[vv] += tt * protoT[off];
      }
    }
  }
  // complete the C-reduction with the pair lane (M split halves)
  ns2 += __shfl_xor(ns2, 16, 32);
  nt2 += __shfl_xor(nt2, 16, 32);
#pragma unroll
  for (int vv = 0; vv < 3; ++vv) {
    dS[vv] += __shfl_xor(dS[vv], 16, 32);
    dT[vv] += __shfl_xor(dT[vv], 16, 32);
  }

  float local = 0.0f;
  if (cok && half == 0) {
    float w  = wbuf[b * HWD + spix];
    float nS = sqrtf(ns2), nT = sqrtf(nt2);
#pragma unroll
    for (int vv = 0; vv < 3; ++vv) {
      float cS = dS[vv] / fmaxf(nS * pnS[(vv * BB + b) * 4 + cls], 1e-8f);
      float cT = dT[vv] / fmaxf(nT * pnT[(vv * BB + b) * 4 + cls], 1e-8f);
      float d  = w * (cS - cT);
      local += d * d;
    }
  }
  for (int off = 16; off > 0; off >>= 1) local += __shfl_xor(local, off, 32);
  if (lane == 0) atomicAdd(lossAcc, local);
}

__global__ void k5_final(const float* lossAcc, float* out) {
  out[0] = lossAcc[0] * (1.0f / (float)(BB * 3 * HWD));
}

extern "C" void kernel_launch(void* const* d_in, const int* in_sizes, int n_in,
                              void* d_out, int out_size, void* d_ws, size_t ws_size,
                              hipStream_t stream) {
  (void)in_sizes; (void)n_in; (void)out_size; (void)ws_size;
  const float* feat_S  = (const float*)d_in[0];
  const float* feat_T  = (const float*)d_in[1];
  const float* Pred_S  = (const float*)d_in[2];
  const float* Pred_T  = (const float*)d_in[3];
  const float* S_atten = (const float*)d_in[4];
  const float* T_atten = (const float*)d_in[5];
  const int*   target  = (const int*)d_in[6];

  float* ws      = (float*)d_ws;
  float* sumTS   = ws + OFF_SUMTS;
  float* sumTTw  = ws + OFF_SUMTT;
  float* cnt     = ws + OFF_CNT;
  float* sct     = ws + OFF_SCT;
  float* protoS  = ws + OFF_PS;
  float* protoT  = ws + OFF_PT;
  float* pnS     = ws + OFF_PNS;
  float* pnT     = ws + OFF_PNT;
  float* lossAcc = ws + OFF_LOSS;
  float* wbuf    = ws + OFF_W;
  float* swbuf   = ws + OFF_SW;
  int*   clsbuf  = (int*)(ws + OFF_CLS);

  kzero<<<(STATS_N + 255) / 256, 256, 0, stream>>>(ws, STATS_N);
  k1_pixels<<<(BB * HWD) / 256, 256, 0, stream>>>(Pred_S, Pred_T, target,
                                                  clsbuf, wbuf, swbuf, cnt, sct);
  k2_stats<<<2048, 256, 0, stream>>>(feat_S, feat_T, S_atten, T_atten,
                                     clsbuf, swbuf, sumTS, sumTTw);
  k3_protos<<<48, 256, 0, stream>>>(sumTS, sumTTw, cnt, sct,
                                    protoS, protoT, pnS, pnT);
  k4_loss<<<896, 256, 0, stream>>>(feat_S, feat_T, S_atten, T_atten,
                                   clsbuf, wbuf, protoS, protoT, pnS, pnT, lossAcc);
  k5_final<<<1, 1, 0, stream>>>(lossAcc, (float*)d_out);
}